// DCModuleOptimized_14998025797937
// MI455X (gfx1250) — compile-verified
//
#include <hip/hip_runtime.h>

namespace {

constexpr int H   = 2047;
constexpr int W   = 2047;
constexpr int NH  = 1023;            // patch grid rows
constexpr int NW  = 1023;            // patch grid cols
constexpr int L   = NH * NW;         // 1046529 = 9 * 116281 (divisible by 9!)
constexpr int LD9 = L / 9;           // 116281
constexpr int BLK = 256;             // 8 wave32s

// One thread per "group" p of the scrambled (L, kk) reshape.
// Group p: fixed patch-element q = p / LD9 (dy=q/3, dx=q%3),
//          locations l = 9*(p % LD9) + j, j = 0..8.
// Element address: img[(2*(l/NW)+dy)*W + 2*(l%NW)+dx].
__global__ __launch_bounds__(BLK)
void triplet_pool(const float* __restrict__ A,
                  const float* __restrict__ P,
                  const float* __restrict__ N,
                  float* __restrict__ out)
{
  // 27 elements per thread staged via async global->LDS engine.
  // Layout sE[slot][tid]: lane-stride-1 readback => bank-conflict free.
  __shared__ float sE[27 * BLK];

  const int tid  = threadIdx.x;
  const int pRaw = blockIdx.x * BLK + tid;
  const int p    = (pRaw < L) ? pRaw : (L - 1);   // clamp: keep EXEC full for loads

  const int q  = p / LD9;            // 0..8
  const int t  = p - q * LD9;
  const int dy = q / 3;
  const int dx = q - dy * 3;
  const int l0 = t * 9;

  int ph = l0 / NW;
  int pw = l0 - ph * NW;

  // LDS byte offset of this lane's slot 0 (flat-pointer low 32 bits == LDS offset).
  const unsigned ldsBase = (unsigned)(uintptr_t)(&sE[0]) + (unsigned)tid * 4u;

  #pragma unroll
  for (int j = 0; j < 9; ++j) {
    const size_t off = (size_t)(2 * ph + dy) * W + (unsigned)(2 * pw + dx);
    const float* ga = A + off;
    const float* gp = P + off;
    const float* gn = N + off;
    const unsigned la = ldsBase + (unsigned)(j * BLK) * 4u;
    const unsigned lp = la + (unsigned)(9 * BLK) * 4u;
    const unsigned ln = la + (unsigned)(18 * BLK) * 4u;
    asm volatile("global_load_async_to_lds_b32 %0, %1, off" :: "v"(la), "v"(ga) : "memory");
    asm volatile("global_load_async_to_lds_b32 %0, %1, off" :: "v"(lp), "v"(gp) : "memory");
    asm volatile("global_load_async_to_lds_b32 %0, %1, off" :: "v"(ln), "v"(gn) : "memory");
    if (++pw == NW) { pw = 0; ++ph; }   // location l crosses a patch-grid row
  }

  // Wait for the async engine to land all 27 elements in this wave's LDS slots.
  asm volatile("s_wait_asynccnt 0" ::: "memory");

  float a[9];
  #pragma unroll
  for (int j = 0; j < 9; ++j) a[j] = sE[j * BLK + tid];

  // First-occurrence argmin/argmax of |a-c| over j (strict compares == jnp tie-break).
  float minDp = 3.402823466e38f, maxDp = -1.0f, minVp = 0.0f, maxVp = 0.0f;
  float minDn = 3.402823466e38f, maxDn = -1.0f, minVn = 0.0f, maxVn = 0.0f;
  #pragma unroll
  for (int j = 0; j < 9; ++j) {
    const float cp = sE[(9 + j) * BLK + tid];
    const float dp = fabsf(a[j] - cp);
    if (dp < minDp) { minDp = dp; minVp = cp; }
    if (dp > maxDp) { maxDp = dp; maxVp = cp; }
    const float cn = sE[(18 + j) * BLK + tid];
    const float dn = fabsf(a[j] - cn);
    if (dn < minDn) { minDn = dn; minVn = cn; }
    if (dn > maxDn) { maxDn = dn; maxVn = cn; }
  }
  const float resP = minVp + maxVp;
  const float resN = minVn + maxVn;

  if (pRaw < L) {
    // Output pixel (i,j) consumes group  min(i/2,1022)*NW + min(j/2,1022):
    // scatter to the 2x2 block (3x3 at last patch row/col).
    const int pi = p / NW;
    const int pj = p - pi * NW;
    const size_t HW = (size_t)H * W;
    float* __restrict__ oP = out;
    float* __restrict__ oN = out + HW;
    const int r0 = 2 * pi, c0 = 2 * pj;
    const int nr = (pi == NH - 1) ? 3 : 2;
    const int nc = (pj == NW - 1) ? 3 : 2;
    for (int i = 0; i < nr; ++i) {
      const size_t ro = (size_t)(r0 + i) * W + (unsigned)c0;
      for (int j = 0; j < nc; ++j) {
        oP[ro + j] = resP;
        oN[ro + j] = resN;
      }
    }
  }
}

} // namespace

extern "C" void kernel_launch(void* const* d_in, const int* in_sizes, int n_in,
                              void* d_out, int out_size, void* d_ws, size_t ws_size,
                              hipStream_t stream)
{
  (void)in_sizes; (void)n_in; (void)out_size; (void)d_ws; (void)ws_size;
  const float* A = (const float*)d_in[0];   // anchor
  const float* P = (const float*)d_in[1];   // positive
  const float* N = (const float*)d_in[2];   // negative
  float* out = (float*)d_out;               // [pos(H*W), neg(H*W)]
  const int blocks = (L + BLK - 1) / BLK;   // 4089
  triplet_pool<<<blocks, BLK, 0, stream>>>(A, P, N, out);
}